// DynamicRouter_52982716563714
// MI455X (gfx1250) — compile-verified
//
#include <hip/hip_runtime.h>
#include <math.h>

// ---------------------------------------------------------------------------
// Dynamic routing (CapsNet) for MI455X / gfx1250.
//
// Bandwidth-bound: 3 mandatory streaming passes over the 256MB inputs tensor
// (sequential dependency through the global n_in reduction each iteration).
// ~768MB traffic -> ~34us roofline at 23.3 TB/s. Compute ~1 GFLOP (trivial).
//
// Structure:
//   pass MODE=0 : S0 partials = sum_i x                    (stream, b128)
//   reduce+squash (WMMA f32 16x16x4 chunk reduction) -> out0
//   pass MODE=1 : L1 = x.out0 ; softmax_j ; S1 partials ; store L1
//   reduce+squash -> out1
//   pass MODE=2 : L2 = L1 + x.out1 ; softmax_j ; S2 partials
//   reduce+squash -> d_out
//
// wave32 layout per row (1024 floats = 256 float4): lane l, slot k:
//   float4 f = row[k*32+l]  ->  j = 8k + l/4 , d = 4*(l%4)..+3
// so per-capsule dot needs only shfl_xor(1),shfl_xor(2); softmax over the 64
// capsules is a full-wave shfl_xor reduction (values 4x replicated -> /4).
// Deterministic: fixed-order shuffles + fixed-order LDS combine, no atomics.
// ---------------------------------------------------------------------------

#define B_SZ    32
#define N_INN   2048
#define J_OUT   64
#define D_DIM   16
#define ROW_F4  256                      // (64*16)/4 float4 per (b,i) row
#define NCHUNK  64                       // partial-sum chunks per batch
#define ROWS_PER_BLOCK (N_INN / NCHUNK)  // 32
#define NWAVES  8
#define ROWS_PER_WAVE (ROWS_PER_BLOCK / NWAVES) // 4
#define EPS_SQ  1e-7f

typedef __attribute__((ext_vector_type(2))) float v2f;
typedef __attribute__((ext_vector_type(8))) float v8f;

__device__ __forceinline__ void f4add(float4& a, const float4 b) {
  a.x += b.x; a.y += b.y; a.z += b.z; a.w += b.w;
}
__device__ __forceinline__ void f4fma(float4& a, float s, const float4 b) {
  a.x = fmaf(s, b.x, a.x); a.y = fmaf(s, b.y, a.y);
  a.z = fmaf(s, b.z, a.z); a.w = fmaf(s, b.w, a.w);
}

// MODE 0: plain sum. MODE 1: softmax(dot) + store logits. MODE 2: softmax(prev+dot).
template <int MODE>
__global__ __launch_bounds__(256) void caps_pass(const float* __restrict__ X,
                                                 const float* __restrict__ Oprev,
                                                 float* __restrict__ Lg,
                                                 float* __restrict__ P) {
  const int chunk = blockIdx.x, b = blockIdx.y;
  const int t = threadIdx.x;
  const int wave = t >> 5, lane = t & 31;
  const int jgrp = lane >> 2;   // capsule group within lane quad
  const int dsub = lane & 3;

  __shared__ float4 ls[NWAVES * ROW_F4];   // 32 KB

  float4 o[8];
  if (MODE != 0) {
    const float4* Ob = (const float4*)Oprev + b * ROW_F4;
#pragma unroll
    for (int k = 0; k < 8; ++k) o[k] = Ob[k * 32 + lane];   // same layout as x
  }

  float4 acc[8];
#pragma unroll
  for (int k = 0; k < 8; ++k) acc[k] = make_float4(0.f, 0.f, 0.f, 0.f);

  const int row0 = chunk * ROWS_PER_BLOCK + wave * ROWS_PER_WAVE;
  const float4* Xb = (const float4*)X + (size_t)b * N_INN * ROW_F4;

  for (int r = 0; r < ROWS_PER_WAVE; ++r) {
    const int i = row0 + r;
    const float4* xr = Xb + (size_t)i * ROW_F4;
    // stream-ahead: each lane prefetches a 128B line covering the next 4KB row
    __builtin_prefetch(xr + ROW_F4 + lane * 8, 0, 0);

    float4 x[8];
#pragma unroll
    for (int k = 0; k < 8; ++k) x[k] = xr[k * 32 + lane];   // global_load_b128

    if (MODE == 0) {
#pragma unroll
      for (int k = 0; k < 8; ++k) f4add(acc[k], x[k]);
    } else {
      float Lv[8];
#pragma unroll
      for (int k = 0; k < 8; ++k) {
        float p = x[k].x * o[k].x + x[k].y * o[k].y + x[k].z * o[k].z + x[k].w * o[k].w;
        p += __shfl_xor(p, 1, 32);          // combine 16 d's across the lane quad
        p += __shfl_xor(p, 2, 32);
        if (MODE == 2)
          p += Lg[((size_t)(b * N_INN) + i) * J_OUT + 8 * k + jgrp];
        Lv[k] = p;                           // logit for j = 8k + jgrp (x4 replicated)
      }
      // softmax over all 64 capsules: full-wave reductions (replication-safe)
      float mx = Lv[0];
#pragma unroll
      for (int k = 1; k < 8; ++k) mx = fmaxf(mx, Lv[k]);
#pragma unroll
      for (int off = 1; off < 32; off <<= 1) mx = fmaxf(mx, __shfl_xor(mx, off, 32));
      float e[8], ss = 0.f;
#pragma unroll
      for (int k = 0; k < 8; ++k) { e[k] = __expf(Lv[k] - mx); ss += e[k]; }
#pragma unroll
      for (int off = 1; off < 32; off <<= 1) ss += __shfl_xor(ss, off, 32);
      const float inv = 4.0f / ss;           // each capsule counted 4x
#pragma unroll
      for (int k = 0; k < 8; ++k) f4fma(acc[k], e[k] * inv, x[k]);

      if (MODE == 1 && dsub == 0) {
#pragma unroll
        for (int k = 0; k < 8; ++k)
          Lg[((size_t)(b * N_INN) + i) * J_OUT + 8 * k + jgrp] = Lv[k];
      }
    }
  }

  // deterministic block combine of the 8 wave partials
#pragma unroll
  for (int k = 0; k < 8; ++k) ls[wave * ROW_F4 + k * 32 + lane] = acc[k];
  __syncthreads();
  float4 s = ls[t];
#pragma unroll
  for (int w = 1; w < NWAVES; ++w) f4add(s, ls[w * ROW_F4 + t]);
  ((float4*)P)[((size_t)(b * NCHUNK) + chunk) * ROW_F4 + t] = s;
}

// Reduce the NCHUNK per-chunk partials with the matrix pipe and apply squash.
// A (16x4 f32) = partials[chunk, e] laid out per the documented A layout:
//   lane L: row m = L&15, VGPR0/1 hold K = 2*(L>>4), 2*(L>>4)+1
// B = ones(4x16)  =>  D[m,n] = sum_k A[m,k]  (exact fp32 chunk reduction).
// C/D layout: lane L, reg v  ->  value S[j*16 + v + 8*(L>>4)].
__global__ __launch_bounds__(256) void caps_reduce_squash(const float* __restrict__ P,
                                                          const float* __restrict__ bias,
                                                          float* __restrict__ O,
                                                          float scale) {
  const int b = blockIdx.x;
  const int t = threadIdx.x, wave = t >> 5, lane = t & 31;
  const int m = lane & 15, half = lane >> 4;
  const v2f ones = {1.0f, 1.0f};

  for (int g = 0; g < J_OUT / NWAVES; ++g) {
    const int j = wave + g * NWAVES;
    v8f acc = {0.f, 0.f, 0.f, 0.f, 0.f, 0.f, 0.f, 0.f};
#pragma unroll
    for (int s = 0; s < NCHUNK / 4; ++s) {
      const int c = 4 * s + 2 * half;
      v2f a;
      a.x = P[((size_t)(b * NCHUNK) + c    ) * 1024 + j * 16 + m];
      a.y = P[((size_t)(b * NCHUNK) + c + 1) * 1024 + j * 16 + m];
      acc = __builtin_amdgcn_wmma_f32_16x16x4_f32(
          /*neg_a=*/false, a, /*neg_b=*/false, ones,
          /*c_mod=*/(short)0, acc, /*reuse_a=*/false, /*reuse_b=*/false);
    }
    // squash(scale*S + bias) over d=16 (split: lanes0-15 d=0..7, lanes16-31 d=8..15)
    float val[8];
    float n2 = 0.f;
#pragma unroll
    for (int v = 0; v < 8; ++v) {
      const int d = v + 8 * half;
      val[v] = scale * acc[v] + bias[j * 16 + d];
      n2 = fmaf(val[v], val[v], n2);
    }
    n2 += __shfl_xor(n2, 16, 32);
    const float coef = (n2 / (1.0f + n2)) / sqrtf(n2 + EPS_SQ);
    if (m == 0) {
#pragma unroll
      for (int v = 0; v < 8; ++v)
        O[(size_t)b * (J_OUT * D_DIM) + j * 16 + 8 * half + v] = val[v] * coef;
    }
  }
}

extern "C" void kernel_launch(void* const* d_in, const int* in_sizes, int n_in,
                              void* d_out, int out_size, void* d_ws, size_t ws_size,
                              hipStream_t stream) {
  const float* X    = (const float*)d_in[0];   // [32,2048,64,16] fp32
  const float* bias = (const float*)d_in[1];   // [64,16] fp32
  float* out = (float*)d_out;                  // [32,64,16] fp32

  // Workspace layout (~24.3 MB total):
  float* P  = (float*)d_ws;                                   // 8 MB partials
  float* O0 = P  + (size_t)B_SZ * NCHUNK * 1024;              // 128 KB
  float* O1 = O0 + B_SZ * J_OUT * D_DIM;                      // 128 KB
  float* Lg = O1 + B_SZ * J_OUT * D_DIM;                      // 16 MB logits

  dim3 grid(NCHUNK, B_SZ);   // 2048 blocks x 8 waves

  caps_pass<0><<<grid, 256, 0, stream>>>(X, nullptr, nullptr, P);
  caps_reduce_squash<<<B_SZ, 256, 0, stream>>>(P, bias, O0, 1.0f / 64.0f);

  caps_pass<1><<<grid, 256, 0, stream>>>(X, O0, Lg, P);
  caps_reduce_squash<<<B_SZ, 256, 0, stream>>>(P, bias, O1, 1.0f);

  caps_pass<2><<<grid, 256, 0, stream>>>(X, O1, Lg, P);
  caps_reduce_squash<<<B_SZ, 256, 0, stream>>>(P, bias, out, 1.0f);
}